// EdgeNetwork_18880676233591
// MI455X (gfx1250) — compile-verified
//
#include <hip/hip_runtime.h>

typedef __attribute__((ext_vector_type(2))) float v2f;
typedef __attribute__((ext_vector_type(8))) float v8f;

#define EN_H  128   // hidden
#define EN_F  64    // edge feature dim
#define EN_ET 64    // edges per workgroup (4 subtiles of 16)

__launch_bounds__(256, 2)
__global__ void edgenet_wmma(const float* __restrict__ nodef,
                             const float* __restrict__ edgef,
                             const int*   __restrict__ edom,
                             const int*   __restrict__ erng,
                             const float* __restrict__ W1,
                             const float* __restrict__ b1,
                             const float* __restrict__ We,
                             float* __restrict__ out,
                             int n_edges)
{
    // x_t[k][e] (k-major, e is LDS bank index) ; region first reused as ef_t[f][e] pad-65
    __shared__ float lds_x[EN_H * EN_ET];
    // mlp_t[j][e]
    __shared__ float lds_m[EN_H * EN_ET];

    const int t    = threadIdx.x;
    const int blk  = blockIdx.x;
    const int e    = t & 63;       // edge-in-tile for stage 1
    const int quad = t >> 6;       // 0..3
    const int eg   = blk * EN_ET + e;

    // ---------- Stage 1a: stage edge_features tile transposed (pad 65) ----------
    {
        const size_t base = (size_t)blk * EN_ET * EN_F;
        #pragma unroll
        for (int i = 0; i < (EN_ET * EN_F) / 256; ++i) {
            const int g  = t + i * 256;
            const int ee = g >> 6;
            const int ff = g & 63;
            float v = (blk * EN_ET + ee < n_edges) ? edgef[base + g] : 0.0f;
            lds_x[ff * 65 + ee] = v;
        }
    }
    __syncthreads();

    // ---------- Stage 1b: edge MLP  mlp[e][j] = relu(ef[e,:]·W1[j,:] + b1[j]) ----------
    {
        const int jb = quad * 32;
        float mv[32];
        for (int jj = 0; jj < 32; ++jj) {
            const int j = jb + jj;
            const float* w = W1 + (size_t)j * EN_F;
            float s = b1[j];
            #pragma unroll 8
            for (int f = 0; f < EN_F; ++f)
                s += lds_x[f * 65 + e] * w[f];
            mv[jj] = fmaxf(s, 0.0f);
        }
        for (int jj = 0; jj < 32; ++jj)
            lds_m[(jb + jj) * EN_ET + e] = mv[jj];
    }
    __syncthreads();  // all reads of ef_t done before overwriting lds_x

    // ---------- Stage 1c: gather x_t[k][e] = nodef[edge_range[e]][k] ----------
    {
        const int r = (eg < n_edges) ? erng[eg] : 0;
        const float* xr = nodef + (size_t)r * EN_H;
        const int kb = quad * 32;
        #pragma unroll 8
        for (int k = 0; k < 32; ++k)
            lds_x[(kb + k) * EN_ET + e] = xr[kb + k];
    }
    __syncthreads();

    // ---------- Stage 2: out_edge = (mlp ⊗ x) · We_r  via V_WMMA_F32_16X16X4_F32 ----------
    // wave wv owns h-block [h0, h0+16); 4 edge subtiles of 16 share each A tile.
    const int lane = t & 31;
    const int wv   = t >> 5;
    const int h0   = wv * 16;
    const int ln   = lane & 15;          // A: h index ; B/D: edge index (N)
    const int hi2  = (lane >> 4) * 2;    // K sub-base for this lane half

    v8f acc[4];
    #pragma unroll
    for (int s = 0; s < 4; ++s) {
        #pragma unroll
        for (int i = 0; i < 8; ++i) acc[s][i] = 0.0f;
    }

    // A[m,ksub] = We[j, (h0+m)*128 + k0 + ksub]  -> lane loads contiguous b64
    const float* Ab = We + (size_t)(h0 + ln) * EN_H + hi2;

    for (int j = 0; j < EN_H; ++j) {
        float ml[4];
        #pragma unroll
        for (int s = 0; s < 4; ++s)                // lanes 16-31 broadcast-read
            ml[s] = lds_m[j * EN_ET + s * 16 + ln];
        const float* Aj = Ab + (size_t)j * (EN_H * EN_H);
        #pragma unroll 4
        for (int k0 = 0; k0 < EN_H; k0 += 4) {
            const v2f a  = *(const v2f*)(Aj + k0);
            const int kk = k0 + hi2;
            #pragma unroll
            for (int s = 0; s < 4; ++s) {
                const int eo = s * 16 + ln;
                const float bx = lds_x[kk * EN_ET + eo];
                const float by = lds_x[(kk + 1) * EN_ET + eo];
                v2f b;
                b.x = ml[s] * bx;   // G[e, j*128+kk]   = mlp[e,j]*x[e,kk]
                b.y = ml[s] * by;   // G[e, j*128+kk+1]
                acc[s] = __builtin_amdgcn_wmma_f32_16x16x4_f32(
                    false, a, false, b, (short)0, acc[s], false, false);
            }
        }
    }

    // ---------- Stage 3: scatter-add into destination nodes ----------
    // D layout: vgpr i, lane L -> h = h0 + i + 8*(L>>4), e = L&15 (per subtile)
    #pragma unroll
    for (int s = 0; s < 4; ++s) {
        const int el = s * 16 + ln;
        const int ge = blk * EN_ET + el;
        if (ge < n_edges) {
            const int d = edom[ge];
            float* op = out + (size_t)d * EN_H + h0 + (lane >> 4) * 8;
            #pragma unroll
            for (int i = 0; i < 8; ++i)
                atomicAdd(op + i, acc[s][i]);
        }
    }
}

extern "C" void kernel_launch(void* const* d_in, const int* in_sizes, int n_in,
                              void* d_out, int out_size, void* d_ws, size_t ws_size,
                              hipStream_t stream) {
    const float* nodef = (const float*)d_in[0];   // [N, 128]
    const float* edgef = (const float*)d_in[1];   // [E, 64]
    const int*   edom  = (const int*)d_in[2];     // [E]
    const int*   erng  = (const int*)d_in[3];     // [E]
    const float* W1    = (const float*)d_in[4];   // [128, 64]
    const float* b1    = (const float*)d_in[5];   // [128]
    const float* We    = (const float*)d_in[6];   // [128, 128*128]
    float* out = (float*)d_out;

    const int n_edges = in_sizes[2];

    // zero the scatter target (harness poisons d_out)
    hipMemsetAsync(out, 0, (size_t)out_size * sizeof(float), stream);

    const int nb = (n_edges + EN_ET - 1) / EN_ET;   // 500 for E=32000
    hipLaunchKernelGGL(edgenet_wmma, dim3(nb), dim3(256), 0, stream,
                       nodef, edgef, edom, erng, W1, b1, We, out, n_edges);
}